// BaselineSAE_34308198760765
// MI455X (gfx1250) — compile-verified
//
#include <hip/hip_runtime.h>

// SAE forward on gfx1250: fp32 WMMA encode (LDS-staged B, double-buffered)
// + radix-select top-k + sparse decode.
typedef __attribute__((ext_vector_type(2))) float v2f;
typedef __attribute__((ext_vector_type(4))) float v4f;
typedef __attribute__((ext_vector_type(8))) float v8f;

#define SAE_IN   256
#define SAE_HID  4096
#define LDSTRIDE 260   // 256 + 4 pad: keeps b128 stores aligned, ds_load_b64
                       // fragment reads hit all 64 LDS banks exactly once.

// ---------------------------------------------------------------------------
// Kernel 1: z = relu((x - b_pre) @ W_enc.T)   via V_WMMA_F32_16X16X4_F32
// grid.x = B/128, 256 threads (8 wave32). Wave w owns rows blk*128+w*16..+15
// with its 16x256 A tile (x - b_pre) resident in 128 VGPRs. All 8 waves sweep
// the same 16-column W tile together; the tile (16x256 f32 = 16KB) is staged
// into LDS cooperatively (coalesced b128) and double-buffered, so W is pulled
// from L2 once per 128 output rows instead of once per 16. B-fragment LDS
// reads are grouped 8-at-a-time into a register buffer so ds_loads pipeline
// (staggered s_wait_dscnt) instead of serializing each WMMA pair.
// ---------------------------------------------------------------------------
__global__ __launch_bounds__(256) void sae_encode_wmma_f32(
    const float* __restrict__ x, const float* __restrict__ Wenc,
    const float* __restrict__ bpre, float* __restrict__ z)
{
  __shared__ float sB[2][16 * LDSTRIDE];

  const int tid  = threadIdx.x;
  const int wave = tid >> 5;
  const int lane = tid & 31;
  const int lh   = lane >> 4;       // 0 -> K pair {4f,4f+1}, 1 -> {4f+2,4f+3}
  const int l16  = lane & 15;

  const int row0 = blockIdx.x * 128 + wave * 16;

  // A fragments per ISA 16x4 f32 layout: lanes 0-15 hold M=l16 K=4f+{0,1},
  // lanes 16-31 hold M=l16 K=4f+{2,3}. Fully unrolled -> stays in VGPRs.
  v2f a[SAE_IN / 4];
  {
    const float* xr = x + (size_t)(row0 + l16) * SAE_IN;
    #pragma unroll
    for (int f = 0; f < SAE_IN / 4; ++f) {
      const int kb = 4 * f + 2 * lh;
      a[f] = *(const v2f*)(xr + kb) - *(const v2f*)(bpre + kb);
    }
  }

  // Cooperative B staging: 16 rows x 256 floats; 4 passes, 4 rows per pass,
  // 64 threads x 16B per row (fully coalesced).
  const int srow = tid >> 6;        // 0..3
  const int scol = (tid & 63) * 4;  // 0..252
  v4f stage[4];

  // Prologue: tile 0 -> sB[0]
  #pragma unroll
  for (int p = 0; p < 4; ++p)
    stage[p] = *(const v4f*)(Wenc + (size_t)(p * 4 + srow) * SAE_IN + scol);
  #pragma unroll
  for (int p = 0; p < 4; ++p)
    *(v4f*)(&sB[0][(p * 4 + srow) * LDSTRIDE + scol]) = stage[p];
  __syncthreads();

  for (int t = 0; t < SAE_HID / 16; ++t) {
    // Issue global loads for tile t+1 while computing tile t.
    if (t + 1 < SAE_HID / 16) {
      const float* wn = Wenc + (size_t)(t + 1) * 16 * SAE_IN;
      #pragma unroll
      for (int p = 0; p < 4; ++p)
        stage[p] = *(const v4f*)(wn + (size_t)(p * 4 + srow) * SAE_IN + scol);
      if (t + 2 < SAE_HID / 16)
        __builtin_prefetch(Wenc + (size_t)(t + 2) * 16 * SAE_IN + tid * 16, 0, 1);
    }

    // 64 WMMAs over K=256 from LDS, in groups of 8: batch the ds_loads into
    // a register buffer so they issue back-to-back and waits stagger.
    // Lane reads sB[l16*260 + 4f + 2lh] (b64): banks (4*l16 + 4f + 2lh) % 64
    // -> all 64 banks exactly once, conflict-free.
    const float* bb = &sB[t & 1][l16 * LDSTRIDE + 2 * lh];
    v8f acc = {};
    #pragma unroll
    for (int g = 0; g < SAE_IN / 4 / 8; ++g) {
      v2f bfr[8];
      #pragma unroll
      for (int u = 0; u < 8; ++u)
        bfr[u] = *(const v2f*)(bb + 4 * (8 * g + u));
      #pragma unroll
      for (int u = 0; u < 8; ++u)
        acc = __builtin_amdgcn_wmma_f32_16x16x4_f32(
            false, a[8 * g + u], false, bfr[u], (short)0, acc, false, false);
    }

    // relu + store: VGPR r, lanes 0-15 -> row row0+r, lanes 16-31 -> row0+8+r.
    {
      float* zp = z + (size_t)(row0 + 8 * lh) * SAE_HID + t * 16 + l16;
      #pragma unroll
      for (int r = 0; r < 8; ++r) {
        float val = acc[r];
        zp[(size_t)r * SAE_HID] = val > 0.0f ? val : 0.0f;
      }
    }

    // Write tile t+1 into the other buffer. Safe with one barrier: the buffer
    // being overwritten was last read in compute(t-1), which completed before
    // the barrier at the end of iteration t-1.
    if (t + 1 < SAE_HID / 16) {
      #pragma unroll
      for (int p = 0; p < 4; ++p)
        *(v4f*)(&sB[(t + 1) & 1][(p * 4 + srow) * LDSTRIDE + scol]) = stage[p];
    }
    __syncthreads();
  }
}

// ---------------------------------------------------------------------------
// Kernel 2: per-row exact top-k (radix select on float bits; relu => keys are
// nonnegative so uint order == float order), z_sparse write, sparse decode.
// One block (256 threads) per row; 16 z-elements per thread in registers.
// ---------------------------------------------------------------------------
__global__ __launch_bounds__(256) void sae_topk_decode(
    const float* __restrict__ z, const float* __restrict__ Wenc,
    const float* __restrict__ bpre, const int* __restrict__ kscalar,
    float* __restrict__ zsp, float* __restrict__ xhat)
{
  const int row = blockIdx.x;
  const int tid = threadIdx.x;
  const int K   = kscalar[0];

  __shared__ int   s_cnt;
  __shared__ int   s_sel;
  __shared__ int   s_scan[256];
  __shared__ float s_val[256];
  __shared__ int   s_idx[256];

  float v[SAE_HID / 256];
  const float* zr = z + (size_t)row * SAE_HID;
  #pragma unroll
  for (int j = 0; j < SAE_HID / 256; ++j) v[j] = zr[tid + 256 * j];

  // MSB-first radix select of the K-th largest key.
  unsigned prefix = 0u;
  for (int bit = 31; bit >= 0; --bit) {
    const unsigned cand = prefix | (1u << bit);
    const unsigned ctop = cand >> bit;
    int c = 0;
    #pragma unroll
    for (int j = 0; j < SAE_HID / 256; ++j)
      c += ((__float_as_uint(v[j]) >> bit) >= ctop) ? 1 : 0;
    #pragma unroll
    for (int m = 16; m >= 1; m >>= 1) c += __shfl_xor(c, m, 32);
    if (tid == 0) s_cnt = 0;
    __syncthreads();
    if ((tid & 31) == 0) atomicAdd(&s_cnt, c);
    __syncthreads();
    if (s_cnt >= K) prefix = cand;
    __syncthreads();
  }

  // Strictly-greater count + tie budget so exactly K survive.
  int cgt = 0, ceq = 0;
  #pragma unroll
  for (int j = 0; j < SAE_HID / 256; ++j) {
    const unsigned key = __float_as_uint(v[j]);
    cgt += (key > prefix) ? 1 : 0;
    ceq += (key == prefix) ? 1 : 0;
  }
  int cg = cgt;
  #pragma unroll
  for (int m = 16; m >= 1; m >>= 1) cg += __shfl_xor(cg, m, 32);
  if (tid == 0) { s_cnt = 0; s_sel = 0; }
  __syncthreads();
  if ((tid & 31) == 0) atomicAdd(&s_cnt, cg);
  s_scan[tid] = ceq;
  __syncthreads();
  const int gt_total = s_cnt;
  for (int off = 1; off < 256; off <<= 1) {   // inclusive scan of eq-counts
    const int add = (tid >= off) ? s_scan[tid - off] : 0;
    __syncthreads();
    s_scan[tid] += add;
    __syncthreads();
  }
  int budget = K - gt_total - (s_scan[tid] - ceq);
  if (budget < 0) budget = 0;
  if (budget > ceq) budget = ceq;

  // Emit full z_sparse row (must materialize zeros) + compact selected list.
  float* zo = zsp + (size_t)row * SAE_HID;
  int taken = 0;
  #pragma unroll
  for (int j = 0; j < SAE_HID / 256; ++j) {
    const unsigned key = __float_as_uint(v[j]);
    bool sel = (key > prefix);
    if (!sel && key == prefix && taken < budget) { sel = true; ++taken; }
    zo[tid + 256 * j] = sel ? v[j] : 0.0f;
    if (sel) {
      const int p = atomicAdd(&s_sel, 1);
      if (p < 256) { s_val[p] = v[j]; s_idx[p] = tid + 256 * j; }
    }
  }
  __syncthreads();
  int nsel = s_sel;
  if (nsel > 256) nsel = 256;

  // Sparse decode: x_hat[row][tid] = b_pre[tid] + sum_j val_j * Wenc[idx_j][tid]
  float acc = bpre[tid];
  for (int j = 0; j < nsel; ++j)
    acc += s_val[j] * Wenc[(size_t)s_idx[j] * SAE_IN + tid];
  xhat[(size_t)row * SAE_IN + tid] = acc;
}

// ---------------------------------------------------------------------------
extern "C" void kernel_launch(void* const* d_in, const int* in_sizes, int n_in,
                              void* d_out, int out_size, void* d_ws, size_t ws_size,
                              hipStream_t stream) {
  (void)n_in; (void)out_size; (void)d_ws; (void)ws_size;
  const float* x    = (const float*)d_in[0];
  const float* Wenc = (const float*)d_in[1];   // (4096,256); W_enc.T == W_dec
  const float* bpre = (const float*)d_in[3];
  const int*   kptr = (const int*)d_in[4];
  const int B = in_sizes[0] / SAE_IN;

  float* xhat = (float*)d_out;                       // (B,256)
  float* z    = xhat + (size_t)B * SAE_IN;           // (B,4096)
  float* zsp  = z    + (size_t)B * SAE_HID;          // (B,4096)

  sae_encode_wmma_f32<<<B / 128, 256, 0, stream>>>(x, Wenc, bpre, z);
  sae_topk_decode<<<B, 256, 0, stream>>>(z, Wenc, bpre, kptr, zsp, xhat);
}